// QNN_49005576848205
// MI455X (gfx1250) — compile-verified
//
#include <hip/hip_runtime.h>
#include <math.h>

// MI455X / gfx1250, wave32.
//
// Closed form of the reference circuit (derived & trace-verified):
//   - All RZ gates are diagonal -> no effect on measured probabilities;
//     CNOT layers are a fixed basis permutation (weights input is dead).
//   - Pre-entangling state is a product state: m_q = cos(x[q]) * cos(x[q+4]).
//   - Composing 6 CNOT-ring layers over GF(2) and inverting:
//       ev = ( m1*m2, m2*m3, m0*m2, m0*m2*m3 )
// Final 4->2 projection via V_WMMA_F32_16X16X4_F32 (exact f32), 2 tiles per
// wave, bias preloaded into the C accumulator.
//
// __cosf -> v_cos_f32: for N(0,1) inputs the reduced-argument rounding error
// is ~2e-6 absolute, negligible vs f32 validation tolerance, and it avoids
// pulling in ocml's huge-argument reduction path (~200 VALU per call).

typedef __attribute__((ext_vector_type(2))) float v2f;
typedef __attribute__((ext_vector_type(8))) float v8f;

#define N_FEAT 12
#define BLOCK  256

__device__ __forceinline__ void qnn_ev(float x0, float x1, float x2, float x3,
                                       float x4, float x5, float x6, float x7,
                                       float ev[4]) {
  const float m0 = __cosf(x0) * __cosf(x4);
  const float m1 = __cosf(x1) * __cosf(x5);
  const float m2 = __cosf(x2) * __cosf(x6);
  const float m3 = __cosf(x3) * __cosf(x7);
  ev[0] = m1 * m2;
  ev[1] = m2 * m3;
  ev[2] = m0 * m2;
  ev[3] = ev[2] * m3;   // m0*m2*m3
}

// ---------------------------------------------------------------------------
// Main kernel: 256 threads = 256 samples per block; 8 waves x 2 WMMA tiles.
// ---------------------------------------------------------------------------
__global__ __launch_bounds__(BLOCK) void qnn_wmma_kernel(
    const float* __restrict__ x, const float* __restrict__ wlin,
    const float* __restrict__ bias, float* __restrict__ out) {
  __shared__ float evs_lds[BLOCK * 4];   // per-sample expectation values
  __shared__ float out_lds[BLOCK * 2];   // per-sample outputs

  const int tid     = threadIdx.x;
  const int gsample = blockIdx.x * BLOCK + tid;

  // Only x[0..7] of each 12-float row are mathematically relevant.
  // Row stride 48 B is 16B-aligned -> two coalesced float4 loads.
  const float4* x4p = (const float4*)(x + (size_t)gsample * N_FEAT);
  const float4 a0 = x4p[0];
  const float4 a1 = x4p[1];

  float ev[4];
  qnn_ev(a0.x, a0.y, a0.z, a0.w, a1.x, a1.y, a1.z, a1.w, ev);

#pragma unroll
  for (int k = 0; k < 4; ++k) evs_lds[tid * 4 + k] = ev[k];

  // Uniform constant-index loads -> s_load broadcasts (no divergent vmem).
  const float sw0 = wlin[0], sw1 = wlin[1], sw2 = wlin[2], sw3 = wlin[3];
  const float sw4 = wlin[4], sw5 = wlin[5], sw6 = wlin[6], sw7 = wlin[7];
  const float sb0 = bias[0], sb1 = bias[1];

  __syncthreads();

  const int lane = tid & 31;
  const int wave = tid >> 5;
  const int half = lane >> 4;   // K-half select (ISA A/B fragment layout)
  const int l16  = lane & 15;

  // B fragment (4x16 f32): column n = l16 (only n<2 real), rows 2h, 2h+1.
  // Branch-free lane selection (v_cndmask) from scalar-held weights.
  const float bx0 = half ? sw2 : sw0;   // n=0: w[0][2h+0]
  const float by0 = half ? sw3 : sw1;   // n=0: w[0][2h+1]
  const float bx1 = half ? sw6 : sw4;   // n=1: w[1][2h+0]
  const float by1 = half ? sw7 : sw5;   // n=1: w[1][2h+1]
  v2f bm;
  bm.x = (l16 == 0) ? bx0 : ((l16 == 1) ? bx1 : 0.0f);
  bm.y = (l16 == 0) ? by0 : ((l16 == 1) ? by1 : 0.0f);
  const float cb = (l16 == 0) ? sb0 : ((l16 == 1) ? sb1 : 0.0f);

#pragma unroll
  for (int tile = 0; tile < 2; ++tile) {
    // A fragment (16x4 f32): row M = l16 of this tile, K = 2*half, 2*half+1
    const int srow = wave * 32 + tile * 16 + l16;
    v2f am;
    am.x = evs_lds[srow * 4 + 2 * half + 0];
    am.y = evs_lds[srow * 4 + 2 * half + 1];

    v8f cacc;
#pragma unroll
    for (int v = 0; v < 8; ++v) cacc[v] = cb;

    // D = A x B + C   (EXEC all-ones here: full block, no early exits)
    v8f dacc = __builtin_amdgcn_wmma_f32_16x16x4_f32(
        false, am, false, bm, (short)0, cacc, false, false);

    // D layout: lane holds column n = l16; VGPR v holds row v + 8*half
    if (l16 < 2) {
#pragma unroll
      for (int v = 0; v < 8; ++v) {
        const int row = wave * 32 + tile * 16 + half * 8 + v;
        out_lds[row * 2 + l16] = dacc[v];
      }
    }
  }
  __syncthreads();

  float2 o;
  o.x = out_lds[tid * 2 + 0];
  o.y = out_lds[tid * 2 + 1];
  ((float2*)out)[gsample] = o;
}

// Scalar tail for B % 256 != 0 (not hit for B=131072, kept for generality).
__global__ void qnn_tail_kernel(const float* __restrict__ x,
                                const float* __restrict__ wlin,
                                const float* __restrict__ bias,
                                float* __restrict__ out, int start, int B) {
  const int s = start + blockIdx.x * blockDim.x + threadIdx.x;
  if (s >= B) return;
  const float* xr = x + (size_t)s * N_FEAT;
  float ev[4];
  qnn_ev(xr[0], xr[1], xr[2], xr[3], xr[4], xr[5], xr[6], xr[7], ev);
#pragma unroll
  for (int n = 0; n < 2; ++n) {
    float acc = bias[n];
#pragma unroll
    for (int k = 0; k < 4; ++k) acc += ev[k] * wlin[n * 4 + k];
    out[(size_t)s * 2 + n] = acc;
  }
}

extern "C" void kernel_launch(void* const* d_in, const int* in_sizes, int n_in,
                              void* d_out, int out_size, void* d_ws, size_t ws_size,
                              hipStream_t stream) {
  const float* x    = (const float*)d_in[0];  // [B,12]
  // d_in[1] (weights) provably does not affect the output (diagonal gates
  // commute with measurement probabilities) -- intentionally unused.
  const float* wlin = (const float*)d_in[2];  // [2,4]
  const float* bias = (const float*)d_in[3];  // [2]
  float* out = (float*)d_out;                 // [B,2]

  const int B     = in_sizes[0] / N_FEAT;
  const int nfull = B / BLOCK;
  const int rem   = B % BLOCK;

  if (nfull > 0)
    qnn_wmma_kernel<<<nfull, BLOCK, 0, stream>>>(x, wlin, bias, out);
  if (rem > 0)
    qnn_tail_kernel<<<1, BLOCK, 0, stream>>>(x, wlin, bias, out,
                                             nfull * BLOCK, B);
}